// SGC_71803263255089
// MI455X (gfx1250) — compile-verified
//
#include <hip/hip_runtime.h>

#define N_NODES   100000
#define N_EDGES   1600000
#define FEAT      64
#define HIDDEN    64
#define NUM_GRAPHS 64
#define BN_EPS    1e-5f

typedef __attribute__((ext_vector_type(2))) float v2f;
typedef __attribute__((ext_vector_type(8))) float v8f;

// ---------------------------------------------------------------- utilities
__global__ void k_fill(float* __restrict__ p, float v, int n) {
    int i = blockIdx.x * blockDim.x + threadIdx.x;
    if (i < n) p[i] = v;
}

// deg[dst] += 1 per edge (deg pre-filled with 1.0 for self loops)
__global__ void k_deg_accum(const int* __restrict__ dst, float* __restrict__ deg, int E) {
    int e = blockIdx.x * blockDim.x + threadIdx.x;
    if (e < E) atomicAdd(&deg[dst[e]], 1.0f);
}

__global__ void k_rsqrt_inplace(float* __restrict__ d, int n) {
    int i = blockIdx.x * blockDim.x + threadIdx.x;
    if (i < n) {
        float v = d[i];
        d[i] = (v > 0.0f) ? rsqrtf(v) : 0.0f;
    }
}

// ---------------------------------------------------------------- propagation
// hout[i,f] = dinv[i]^2 * hin[i,f]   (self-loop term; also zero-initializes hout)
__global__ void k_hop_init(const float* __restrict__ hin, const float* __restrict__ dinv,
                           float* __restrict__ hout, int n) {
    int i = blockIdx.x * blockDim.x + threadIdx.x;
    if (i < n) {
        int node = i >> 6;
        float di = dinv[node];
        hout[i] = di * di * hin[i];
    }
}

// one wave per edge: lane covers features {lane, lane+32}
__global__ __launch_bounds__(256) void k_hop_edges(
    const int* __restrict__ src, const int* __restrict__ dst,
    const float* __restrict__ dinv, const float* __restrict__ hin,
    float* __restrict__ hout, int E) {
    int wave = blockIdx.x * (blockDim.x >> 5) + (threadIdx.x >> 5);
    if (wave >= E) return;
    int lane = threadIdx.x & 31;
    int s = src[wave];
    int d = dst[wave];
    float norm = dinv[s] * dinv[d];
    const float* hi = hin + (size_t)s * 64;
    float*       ho = hout + (size_t)d * 64;
    atomicAdd(&ho[lane],      norm * hi[lane]);
    atomicAdd(&ho[lane + 32], norm * hi[lane + 32]);
}

// ---------------------------------------------------------------- linear+relu (WMMA fp32)
// Hout = relu(H @ W^T + b)  with H [N,64], W [64,64] row-major (out col j uses W[j][:]).
// One wave per 16-row tile; 4 col tiles of 16; K=64 in 16 steps of K=4.
__global__ __launch_bounds__(256) void k_linear_relu_wmma(
    const float* __restrict__ H, const float* __restrict__ W,
    const float* __restrict__ bias, float* __restrict__ Hout, int nRowTiles) {
    int wave = blockIdx.x * (blockDim.x >> 5) + (threadIdx.x >> 5);
    if (wave >= nRowTiles) return;           // whole-wave exit: EXEC stays all-ones
    int lane = threadIdx.x & 31;
    int lo = lane & 15;                       // A: row M; B: col N; D: col N
    int hi = lane >> 4;                       // K sub-select (0 -> K{0,1}, 1 -> K{2,3})

    const float* arow = H + ((size_t)(wave * 16 + lo)) * 64 + 2 * hi;
    const float* wr0 = W + ((size_t)(0 * 16 + lo)) * 64 + 2 * hi;
    const float* wr1 = W + ((size_t)(1 * 16 + lo)) * 64 + 2 * hi;
    const float* wr2 = W + ((size_t)(2 * 16 + lo)) * 64 + 2 * hi;
    const float* wr3 = W + ((size_t)(3 * 16 + lo)) * 64 + 2 * hi;

    v8f acc0 = {}, acc1 = {}, acc2 = {}, acc3 = {};
#pragma unroll
    for (int kk = 0; kk < 16; ++kk) {
        int k = kk * 4;
        v2f a;  a.x = arow[k]; a.y = arow[k + 1];
        v2f b0; b0.x = wr0[k]; b0.y = wr0[k + 1];
        v2f b1; b1.x = wr1[k]; b1.y = wr1[k + 1];
        v2f b2; b2.x = wr2[k]; b2.y = wr2[k + 1];
        v2f b3; b3.x = wr3[k]; b3.y = wr3[k + 1];
        acc0 = __builtin_amdgcn_wmma_f32_16x16x4_f32(false, a, false, b0, (short)0, acc0, false, false);
        acc1 = __builtin_amdgcn_wmma_f32_16x16x4_f32(false, a, false, b1, (short)0, acc1, false, false);
        acc2 = __builtin_amdgcn_wmma_f32_16x16x4_f32(false, a, false, b2, (short)0, acc2, false, false);
        acc3 = __builtin_amdgcn_wmma_f32_16x16x4_f32(false, a, false, b3, (short)0, acc3, false, false);
    }

    float bb0 = bias[0 * 16 + lo];
    float bb1 = bias[1 * 16 + lo];
    float bb2 = bias[2 * 16 + lo];
    float bb3 = bias[3 * 16 + lo];
#pragma unroll
    for (int r = 0; r < 8; ++r) {
        int orow = wave * 16 + r + 8 * hi;    // D: VGPR r holds M = r + 8*hi
        float* out = Hout + (size_t)orow * 64;
        out[0 * 16 + lo] = fmaxf(acc0[r] + bb0, 0.0f);
        out[1 * 16 + lo] = fmaxf(acc1[r] + bb1, 0.0f);
        out[2 * 16 + lo] = fmaxf(acc2[r] + bb2, 0.0f);
        out[3 * 16 + lo] = fmaxf(acc3[r] + bb3, 0.0f);
    }
}

// ---------------------------------------------------------------- batchnorm stats
__global__ __launch_bounds__(256) void k_bn_stats(const float* __restrict__ H,
                                                  float* __restrict__ gsum,
                                                  float* __restrict__ gsumsq) {
    __shared__ float s_s[256];
    __shared__ float s_q[256];
    int t = threadIdx.x;
    int f = t & 63;
    int g = t >> 6;                           // 0..3 row-groups per block
    float s = 0.0f, q = 0.0f;
    for (int row = blockIdx.x * 4 + g; row < N_NODES; row += gridDim.x * 4) {
        float v = H[(size_t)row * 64 + f];
        s += v;
        q += v * v;
    }
    s_s[t] = s; s_q[t] = q;
    __syncthreads();
    if (t < 64) {
        s = s_s[t] + s_s[t + 64] + s_s[t + 128] + s_s[t + 192];
        q = s_q[t] + s_q[t + 64] + s_q[t + 128] + s_q[t + 192];
        atomicAdd(&gsum[f], s);
        atomicAdd(&gsumsq[f], q);
    }
}

__global__ void k_bn_finalize(const float* __restrict__ gsum, const float* __restrict__ gsumsq,
                              const float* __restrict__ gamma, const float* __restrict__ beta,
                              float* __restrict__ scale, float* __restrict__ shift) {
    int f = threadIdx.x;
    if (f < HIDDEN) {
        const float invN = 1.0f / (float)N_NODES;
        float mean = gsum[f] * invN;
        float var  = gsumsq[f] * invN - mean * mean;
        float inv  = rsqrtf(var + BN_EPS);
        float sc   = gamma[f] * inv;
        scale[f] = sc;
        shift[f] = beta[f] - mean * sc;
    }
}

// y = h*scale + shift; write back in place; pool by graph id (batch is sorted)
__global__ void k_bn_apply_pool(float* __restrict__ H, const int* __restrict__ batch,
                                const float* __restrict__ scale, const float* __restrict__ shift,
                                float* __restrict__ pooled, int n) {
    int i = blockIdx.x * blockDim.x + threadIdx.x;
    if (i < n) {
        int f = i & 63;
        int node = i >> 6;
        float y = H[i] * scale[f] + shift[f];
        H[i] = y;
        atomicAdd(&pooled[batch[node] * 64 + f], y);
    }
}

// ---------------------------------------------------------------- launcher
extern "C" void kernel_launch(void* const* d_in, const int* in_sizes, int n_in,
                              void* d_out, int out_size, void* d_ws, size_t ws_size,
                              hipStream_t stream) {
    const float* x     = (const float*)d_in[0];                 // [N,64]
    const int*   edge  = (const int*)d_in[1];                   // [2,E]
    const int*   batch = (const int*)d_in[2];                   // [N]
    const float* W     = (const float*)d_in[3];                 // [64,64]
    const float* bias  = (const float*)d_in[4];                 // [64]
    const float* gamma = (const float*)d_in[5];                 // [64]
    const float* beta  = (const float*)d_in[6];                 // [64]

    float* out    = (float*)d_out;
    float* pooled = out;                                        // [64,64]
    float* outH   = out + NUM_GRAPHS * HIDDEN;                  // [N,64] (also ping-pong buf)

    float* ws    = (float*)d_ws;                                // ~26 MB used
    float* bufA  = ws;                                          // [N,64]
    float* dinv  = bufA + (size_t)N_NODES * 64;                 // [N] (deg, then rsqrt in place)
    float* gsum  = dinv + N_NODES;                              // [64]
    float* gsq   = gsum + 64;                                   // [64]
    float* scale = gsq + 64;                                    // [64]
    float* shift = scale + 64;                                  // [64]

    const int* src = edge;
    const int* dst = edge + N_EDGES;

    const int NF = N_NODES * FEAT;                              // 6.4M
    const int TB = 256;

    // degree (self loops give deg=1 baseline) -> dinv = rsqrt(deg)
    k_fill<<<(N_NODES + TB - 1) / TB, TB, 0, stream>>>(dinv, 1.0f, N_NODES);
    k_deg_accum<<<(N_EDGES + TB - 1) / TB, TB, 0, stream>>>(dst, dinv, N_EDGES);
    k_rsqrt_inplace<<<(N_NODES + TB - 1) / TB, TB, 0, stream>>>(dinv, N_NODES);

    const int edgeBlocks = (N_EDGES + 7) / 8;                   // 8 waves/block, 1 wave/edge
    const int nfBlocks   = (NF + TB - 1) / TB;

    // hop 1: x -> bufA
    k_hop_init <<<nfBlocks, TB, 0, stream>>>(x, dinv, bufA, NF);
    k_hop_edges<<<edgeBlocks, TB, 0, stream>>>(src, dst, dinv, x, bufA, N_EDGES);
    // hop 2: bufA -> outH
    k_hop_init <<<nfBlocks, TB, 0, stream>>>(bufA, dinv, outH, NF);
    k_hop_edges<<<edgeBlocks, TB, 0, stream>>>(src, dst, dinv, bufA, outH, N_EDGES);
    // hop 3: outH -> bufA
    k_hop_init <<<nfBlocks, TB, 0, stream>>>(outH, dinv, bufA, NF);
    k_hop_edges<<<edgeBlocks, TB, 0, stream>>>(src, dst, dinv, outH, bufA, N_EDGES);

    // linear + relu via fp32 WMMA: bufA -> outH
    const int rowTiles = N_NODES / 16;                          // 6250 exactly
    k_linear_relu_wmma<<<(rowTiles + 7) / 8, TB, 0, stream>>>(bufA, W, bias, outH, rowTiles);

    // batchnorm + pool
    k_fill<<<(NUM_GRAPHS * HIDDEN + TB - 1) / TB, TB, 0, stream>>>(pooled, 0.0f, NUM_GRAPHS * HIDDEN);
    k_fill<<<1, 128, 0, stream>>>(gsum, 0.0f, 128);             // gsum+gsq contiguous
    k_bn_stats<<<512, TB, 0, stream>>>(outH, gsum, gsq);
    k_bn_finalize<<<1, 64, 0, stream>>>(gsum, gsq, gamma, beta, scale, shift);
    k_bn_apply_pool<<<nfBlocks, TB, 0, stream>>>(outH, batch, scale, shift, pooled, NF);
}